// GCN_62491774157400
// MI455X (gfx1250) — compile-verified
//
#include <hip/hip_runtime.h>
#include <math.h>

// The Python harness owns main(), I/O, device malloc/memcpy, and timing.

typedef float v2f __attribute__((ext_vector_type(2)));
typedef float v8f __attribute__((ext_vector_type(8)));

#define BN_EPS 1e-5f

// ---------------- utility kernels ----------------

__global__ void k_fill(float* __restrict__ p, float val, unsigned n) {
  unsigned i = blockIdx.x * blockDim.x + threadIdx.x;
  if (i < n) p[i] = val;
}

__global__ void k_count_deg(const long long* __restrict__ dst,
                            float* __restrict__ deg, unsigned E) {
  unsigned e = blockIdx.x * blockDim.x + threadIdx.x;
  if (e < E) atomicAdd(&deg[(int)dst[e]], 1.0f);
}

__global__ void k_deg_finalize(const float* __restrict__ deg,
                               float* __restrict__ dis, float* __restrict__ di,
                               unsigned n) {
  unsigned i = blockIdx.x * blockDim.x + threadIdx.x;
  if (i < n) {
    float d = deg[i];
    dis[i] = rsqrtf(d);
    di[i] = 1.0f / d;
  }
}

// ---------------- fp32 WMMA GEMM: H[N,dout] = X[N,din] @ W[din,dout] ----------------
// One wave computes TNW adjacent 16x16 output tiles along N, sharing a single
// A fragment per k-step (A loads are 512B-strided across lanes -> amortize them).
// V_WMMA_F32_16X16X4_F32, K-step 4.
//   A 16x4 f32: lane l -> M=l&15; v0 holds K=k+2*(l>>4), v1 holds K+1.
//   B 4x16 f32: mirrored row-striping across lanes.
//   C/D: vgpr r, lane l -> M = r + 8*(l>>4), N = l&15.
template <int TNW>
__global__ void k_gemm_wmma(const float* __restrict__ X, const float* __restrict__ W,
                            float* __restrict__ H, int din, int dout,
                            int groupsN, int totalGroups) {
  int wave = (int)((blockIdx.x * blockDim.x + threadIdx.x) >> 5);
  if (wave >= totalGroups) return;  // wave-uniform exit: EXEC stays all-1s for WMMA
  int lane = threadIdx.x & 31;
  int half = lane >> 4;     // 0: lanes 0-15, 1: lanes 16-31
  int m16  = lane & 15;
  int tm = wave / groupsN;
  int tg = wave - tm * groupsN;
  int row = tm * 16 + m16;                // A-matrix row for this lane
  int colBase = tg * (16 * TNW) + m16;    // first B/C/D column for this lane

  const float* __restrict__ xr = X + (size_t)row * din;
  v8f acc[TNW];
#pragma unroll
  for (int j = 0; j < TNW; j++) acc[j] = (v8f){0.f, 0.f, 0.f, 0.f, 0.f, 0.f, 0.f, 0.f};

  for (int k = 0; k < din; k += 4) {
    int kk = k + half * 2;
    v2f a;
    a.x = xr[kk];
    a.y = xr[kk + 1];                     // contiguous float2 per lane
    const float* __restrict__ w0 = W + (size_t)kk * dout;
#pragma unroll
    for (int j = 0; j < TNW; j++) {
      v2f b;
      b.x = w0[colBase + j * 16];
      b.y = w0[dout + colBase + j * 16];
      acc[j] = __builtin_amdgcn_wmma_f32_16x16x4_f32(
          /*neg_a=*/false, a, /*neg_b=*/false, b,
          /*c_mod=*/(short)0, acc[j], /*reuse_a=*/false, /*reuse_b=*/false);
    }
  }

  float* __restrict__ hr = H + (size_t)(tm * 16 + half * 8) * dout + colBase;
#pragma unroll
  for (int j = 0; j < TNW; j++)
#pragma unroll
    for (int r = 0; r < 8; r++) hr[(size_t)r * dout + j * 16] = acc[j][r];
}

// ---------------- edge scatter: out[dst] += dis[src]*dis[dst] * h[src] ----------------
// One wave32 per (edge, 32-channel chunk): edge indices loaded once per 32
// channels; gathers/atomics are coalesced 128B row segments.
__global__ void k_scatter(const long long* __restrict__ src,
                          const long long* __restrict__ dst,
                          const float* __restrict__ dis,
                          const float* __restrict__ h,
                          float* __restrict__ out,
                          unsigned nWaves, int chunkShift, int shift) {
  unsigned gw = (blockIdx.x * blockDim.x + threadIdx.x) >> 5;
  if (gw >= nWaves) return;
  int lane = threadIdx.x & 31;
  unsigned e = gw >> chunkShift;
  unsigned chunk = gw & ((1u << chunkShift) - 1u);
  int c = (int)(chunk << 5) + lane;
  int s = (int)src[e];
  int d = (int)dst[e];
  float w = dis[s] * dis[d];
  atomicAdd(&out[((size_t)d << shift) + c], w * h[((size_t)s << shift) + c]);
}

// out += di[n] * h + bias[c]   (self-loop term + conv bias, fused)
__global__ void k_self_bias(float* __restrict__ out, const float* __restrict__ h,
                            const float* __restrict__ di, const float* __restrict__ b,
                            unsigned total, int shift, int mask) {
  unsigned i = blockIdx.x * blockDim.x + threadIdx.x;
  if (i >= total) return;
  unsigned n = i >> shift;
  int c = i & mask;
  out[i] += di[n] * h[i] + b[c];
}

// ---------------- BatchNorm stats: per-channel sum & sum-of-squares ----------------
// blockDim=256 is a multiple of every dout (32..256), so each thread's channel is
// fixed across its grid-stride loop. LDS reduce per block, then one atomic per
// channel per block.
__global__ void k_bn_stats(const float* __restrict__ v, float* __restrict__ sum,
                           float* __restrict__ sumsq, unsigned total, int mask) {
  __shared__ float ss[256];
  __shared__ float sq[256];
  unsigned tid = threadIdx.x;
  unsigned idx = blockIdx.x * blockDim.x + tid;
  unsigned stride = gridDim.x * blockDim.x;
  float s = 0.f, q = 0.f;
  for (unsigned i = idx; i < total; i += stride) {
    float t = v[i];
    s += t;
    q += t * t;
  }
  ss[tid] = s;
  sq[tid] = q;
  __syncthreads();
  int dout = mask + 1;
  if ((int)tid < dout) {
    float a = 0.f, bsum = 0.f;
    for (int j = tid; j < 256; j += dout) {
      a += ss[j];
      bsum += sq[j];
    }
    atomicAdd(&sum[tid], a);
    atomicAdd(&sumsq[tid], bsum);
  }
}

// x_out = relu((v - mu) * rsqrt(var + eps) * g + be)
__global__ void k_bn_apply(const float* __restrict__ v, float* __restrict__ xo,
                           const float* __restrict__ sum, const float* __restrict__ sumsq,
                           const float* __restrict__ g, const float* __restrict__ be,
                           float Nf, unsigned total, int mask) {
  unsigned i = blockIdx.x * blockDim.x + threadIdx.x;
  if (i >= total) return;
  int c = i & mask;
  float mu = sum[c] / Nf;
  float var = sumsq[c] / Nf - mu * mu;
  float t = (v[i] - mu) * rsqrtf(var + BN_EPS) * g[c] + be[c];
  xo[i] = t > 0.f ? t : 0.f;
}

// ---------------- edge head: sigmoid((x[src]*x[dst]) @ fc_w + fc_b) ----------------
// One wave32 per edge; lane == channel (final dim is exactly 32). Shuffle-reduce.
__global__ void k_edge_head(const long long* __restrict__ src,
                            const long long* __restrict__ dst,
                            const float* __restrict__ x,
                            const float* __restrict__ fcw,
                            const float* __restrict__ fcb,
                            float* __restrict__ out, unsigned E) {
  unsigned wave = (blockIdx.x * blockDim.x + threadIdx.x) >> 5;
  if (wave >= E) return;
  int lane = threadIdx.x & 31;
  int s = (int)src[wave];
  int d = (int)dst[wave];
  float v = x[(size_t)s * 32 + lane] * x[(size_t)d * 32 + lane] * fcw[lane];
#pragma unroll
  for (int off = 16; off > 0; off >>= 1) v += __shfl_xor(v, off, 32);
  if (lane == 0) out[wave] = 1.0f / (1.0f + expf(-(v + fcb[0])));
}

// ---------------- orchestration ----------------

extern "C" void kernel_launch(void* const* d_in, const int* in_sizes, int n_in,
                              void* d_out, int out_size, void* d_ws, size_t ws_size,
                              hipStream_t stream) {
  const float* x_in = (const float*)d_in[0];
  const long long* ei = (const long long*)d_in[1];  // int64 per reference
  const int N = in_sizes[0] / 128;
  const unsigned E = (unsigned)(in_sizes[1] / 2);
  const long long* src = ei;
  const long long* dst = ei + E;

  const float *Ws[5], *bs[5], *gs[5], *bes[5];
  for (int i = 0; i < 5; i++) {
    Ws[i]  = (const float*)d_in[2 + 4 * i];
    bs[i]  = (const float*)d_in[3 + 4 * i];
    gs[i]  = (const float*)d_in[4 + 4 * i];
    bes[i] = (const float*)d_in[5 + 4 * i];
  }
  const float* fcw = (const float*)d_in[22];
  const float* fcb = (const float*)d_in[23];
  float* out = (float*)d_out;

  // workspace carve-up
  float* ws = (float*)d_ws;
  float* deg = ws;                 // N
  float* dis = ws + N;             // N
  float* di  = ws + 2 * (size_t)N; // N
  float* sum = ws + 3 * (size_t)N; // 256
  float* sumsq = sum + 256;        // 256
  float* base = ws + 3 * (size_t)N + 512;
  const size_t NB = (size_t)N * 256;
  float* bufs[3] = {base, base + NB, base + 2 * NB};

  const int dims[6] = {128, 128, 256, 128, 64, 32};
  const int B = 256;

  // degrees (with self-loop): deg = 1 + in-degree; dis = deg^-1/2; di = deg^-1
  k_fill<<<(N + B - 1) / B, B, 0, stream>>>(deg, 1.0f, (unsigned)N);
  k_count_deg<<<(E + B - 1) / B, B, 0, stream>>>(dst, deg, E);
  k_deg_finalize<<<(N + B - 1) / B, B, 0, stream>>>(deg, dis, di, (unsigned)N);

  const float* xc = x_in;
  int hb = 0;
  for (int i = 0; i < 5; i++) {
    const int din = dims[i], dout = dims[i + 1];
    hb = i % 3;
    const int cb = (i + 1) % 3;
    float* h = bufs[hb];
    float* conv = bufs[cb];

    // GEMM: TNW=4 N-tiles per wave when dout>=64, else TNW=2 (dout==32).
    const int tilesM = N / 16, tilesN = dout / 16;
    if (dout >= 64) {
      const int groupsN = tilesN / 4;
      const int totalGroups = tilesM * groupsN;
      k_gemm_wmma<4><<<(totalGroups * 32 + 127) / 128, 128, 0, stream>>>(
          xc, Ws[i], h, din, dout, groupsN, totalGroups);
    } else {
      const int groupsN = tilesN / 2;
      const int totalGroups = tilesM * groupsN;
      k_gemm_wmma<2><<<(totalGroups * 32 + 127) / 128, 128, 0, stream>>>(
          xc, Ws[i], h, din, dout, groupsN, totalGroups);
    }

    const unsigned total = (unsigned)N * (unsigned)dout;
    const int shift = __builtin_ctz((unsigned)dout);
    const int mask = dout - 1;
    const int chunkShift = shift - 5;  // log2(dout/32), dout >= 32 always

    k_fill<<<(total + B - 1) / B, B, 0, stream>>>(conv, 0.0f, total);
    const unsigned nWaves = E << chunkShift;  // E * dout/32, max 25.6M
    k_scatter<<<(nWaves * 32u + 255u) / 256u, 256, 0, stream>>>(
        src, dst, dis, h, conv, nWaves, chunkShift, shift);
    k_self_bias<<<(total + B - 1) / B, B, 0, stream>>>(conv, h, di, bs[i], total,
                                                       shift, mask);
    k_fill<<<2, B, 0, stream>>>(sum, 0.0f, 512u);
    k_bn_stats<<<1024, 256, 0, stream>>>(conv, sum, sumsq, total, mask);
    // write x_next into the h buffer (h is dead after self_bias)
    k_bn_apply<<<(total + B - 1) / B, B, 0, stream>>>(conv, h, sum, sumsq, gs[i],
                                                      bes[i], (float)N, total, mask);
    xc = h;
  }

  // edge head: one wave32 per edge (last hidden dim == 32 == wave width)
  k_edge_head<<<(E * 32 + 255) / 256, 256, 0, stream>>>(src, dst, xc, fcw, fcb,
                                                        out, E);
}